// MedianPool2d_24670292148387
// MI455X (gfx1250) — compile-verified
//
#include <hip/hip_runtime.h>

// 3x3 median pool, stride 1, reflect pad 1, on [N=16, C=3, H=512, W=512] fp32.
//
// MI455X strategy:
//  - Memory-bound: ~100MB traffic -> ~4.3us floor @ 23.3 TB/s.
//  - Stage an 18x66 halo tile per block into LDS via the CDNA5 async
//    Memory->LDS path (global_load_async_to_lds_b32 / ASYNCcnt); reflect
//    padding resolved once during staging.
//  - Median-of-9 via the sort-columns identity:
//        med9 = med3( max3(col mins), med3(col meds), min3(col maxs) )
//    lowering to hardware v_min3/v_med3/v_max3 (~8.5 VALU ops / output).
//  - Each thread produces 4 horizontally adjacent outputs: LDS reads are
//    float4+float2 per row (3 rows -> 6 ds ops), output is one b128 store.

#define TILE_W 64
#define TILE_H 16
#define LROWS  (TILE_H + 2)   // 18 rows staged (halo)
#define LSTR   (TILE_W + 4)   // 68 floats/row: 66 used + pad; 272B = 16B-aligned
#define NSTAGE (LROWS * LSTR) // 1224 elements staged per block
#define NTHREADS 256

__device__ __forceinline__ int reflect_idx(int i, int n) {
    i = (i < 0) ? -i : i;               // -1 -> 1
    i = (i >= n) ? (2 * n - 2 - i) : i; // n -> n-2, n+k -> n-2-k
    return i;
}

__device__ __forceinline__ float min3f(float a, float b, float c) {
    return fminf(fminf(a, b), c);            // -> v_min3_num_f32
}
__device__ __forceinline__ float max3f(float a, float b, float c) {
    return fmaxf(fmaxf(a, b), c);            // -> v_max3_num_f32
}
__device__ __forceinline__ float med3f(float a, float b, float c) {
    return __builtin_amdgcn_fmed3f(a, b, c); // -> v_med3_num_f32
}

typedef __attribute__((address_space(3))) float lds_float;

__global__ __launch_bounds__(NTHREADS)
void MedianPool2d_24670292148387_kernel(const float* __restrict__ x,
                                        float* __restrict__ out,
                                        int H, int W) {
    __shared__ float tile[NSTAGE];

    const int tid = threadIdx.x;
    const int x0  = blockIdx.x * TILE_W;
    const int y0  = blockIdx.y * TILE_H;
    const size_t plane = (size_t)blockIdx.z * (size_t)H * (size_t)W;
    const float* img = x + plane;

    // addrspace(3) pointer value == byte offset within the block's LDS
    lds_float* t3 = (lds_float*)tile;

    // ---- Stage halo tile (rows y0-1..y0+16, cols x0-1..x0+66) into LDS
    //      via async Memory->LDS DMA; reflect padding applied here. ----
    for (int e = tid; e < NSTAGE; e += NTHREADS) {
        const int r  = e / LSTR;
        const int c  = e - r * LSTR;
        const int gy = reflect_idx(y0 - 1 + r, H);
        const int gx = reflect_idx(x0 - 1 + c, W);
        const float* gp = img + (size_t)gy * (size_t)W + (size_t)gx;
        const unsigned lds_off = (unsigned)(unsigned long long)(t3 + e);
        asm volatile("global_load_async_to_lds_b32 %0, %1, off"
                     :: "v"(lds_off), "v"(gp)
                     : "memory");
    }
    asm volatile("s_wait_asynccnt 0" ::: "memory");
    __syncthreads();

    // ---- Compute: thread (tx, ty) produces outputs at tile row ty,
    //      cols 4*tx .. 4*tx+3. LDS row k = global row y0-1+k,
    //      LDS col j = global col x0-1+j. Output (ty, c) needs LDS rows
    //      ty..ty+2, cols c..c+2; this thread needs cols 4tx..4tx+5. ----
    const int tx = tid & 15;  // 0..15 -> column quad
    const int ty = tid >> 4;  // 0..15 -> output row within tile
    const int cb = tx << 2;   // 4tx: 16B-aligned LDS column base

    float r0[6], r1[6], r2[6];
#pragma unroll
    for (int k = 0; k < 3; ++k) {
        const float* p = &tile[(ty + k) * LSTR + cb];
        const float4 q = *(const float4*)(p);       // ds_load_b128
        const float2 d = *(const float2*)(p + 4);   // ds_load_b64
        float* dst = (k == 0) ? r0 : (k == 1) ? r1 : r2;
        dst[0] = q.x; dst[1] = q.y; dst[2] = q.z; dst[3] = q.w;
        dst[4] = d.x; dst[5] = d.y;
    }

    float cmn[6], cmd[6], cmx[6];
#pragma unroll
    for (int j = 0; j < 6; ++j) {
        cmn[j] = min3f(r0[j], r1[j], r2[j]);
        cmd[j] = med3f(r0[j], r1[j], r2[j]);
        cmx[j] = max3f(r0[j], r1[j], r2[j]);
    }

    float4 res;
    float* rp = (float*)&res;
#pragma unroll
    for (int i = 0; i < 4; ++i) {
        rp[i] = med3f(max3f(cmn[i], cmn[i + 1], cmn[i + 2]),
                      med3f(cmd[i], cmd[i + 1], cmd[i + 2]),
                      min3f(cmx[i], cmx[i + 1], cmx[i + 2]));
    }

    const int oy = y0 + ty;
    const int ox = x0 + cb;
    if (oy < H && ox + 3 < W) {
        float* o = out + plane + (size_t)oy * (size_t)W + (size_t)ox;
        *(float4*)o = res;                          // global_store_b128
    }
}

extern "C" void kernel_launch(void* const* d_in, const int* in_sizes, int n_in,
                              void* d_out, int out_size, void* d_ws, size_t ws_size,
                              hipStream_t stream) {
    (void)n_in; (void)d_ws; (void)ws_size; (void)out_size;
    const float* x = (const float*)d_in[0];
    float* out = (float*)d_out;

    const int H = 512, W = 512;
    const int nplanes = in_sizes[0] / (H * W); // N*C = 48

    dim3 grid((W + TILE_W - 1) / TILE_W,   // 8
              (H + TILE_H - 1) / TILE_H,   // 32
              nplanes);                    // 48
    dim3 block(NTHREADS);
    MedianPool2d_24670292148387_kernel<<<grid, block, 0, stream>>>(x, out, H, W);
}